// RGCNLayer_43155831390586
// MI455X (gfx1250) — compile-verified
//
#include <hip/hip_runtime.h>

// ---------------------------------------------------------------------------
// RGCN layer for MI455X (gfx1250), fp32 end-to-end via V_WMMA_F32_16X16X4_F32.
// ---------------------------------------------------------------------------

typedef float v2f __attribute__((ext_vector_type(2)));
typedef float v8f __attribute__((ext_vector_type(8)));

#define DIM 128
#define RRELU_SLOPE 0.22916666666666666f  // (1/8 + 1/3)/2

// ---------------------------------------------------------------------------
__global__ __launch_bounds__(256) void rgcn_zero_ws(float* __restrict__ p, long n) {
    long i = (long)blockIdx.x * blockDim.x + threadIdx.x;
    long stride = (long)gridDim.x * blockDim.x;
    for (; i < n; i += stride) p[i] = 0.0f;
}

// ---------------------------------------------------------------------------
// One wave handles 16 edges: message = (nodes[src] + rels[rel]) @ W_agg^T,
// then atomic scatter-add into summed[dst] and count[dst].
// ---------------------------------------------------------------------------
__global__ __launch_bounds__(128) void rgcn_edge_kernel(
        const float* __restrict__ nodes,   // n_nodes x 128
        const float* __restrict__ rels,    // n_rel   x 128
        const float* __restrict__ W_agg,   // 128 x 128
        const int*   __restrict__ edges,   // n_edges x 3 (src, rel, dst)
        float* __restrict__ summed,        // n_nodes x 128 (accumulator)
        float* __restrict__ count,         // n_nodes
        int n_edges) {
    __shared__ float xs[4][16][DIM];   // per-wave 16x128 staged A rows (32 KB)
    __shared__ int   dsts[4][16];

    const int wave = threadIdx.x >> 5;
    const int lane = threadIdx.x & 31;
    const int l15  = lane & 15;
    const int half = lane >> 4;        // 0 or 1

    const int ebase = (blockIdx.x * 4 + wave) * 16;
    if (ebase >= n_edges) return;      // uniform across the wave

    // ---- stage x = nodes[src] + rels[rel] into wave-private LDS -----------
    for (int r = 0; r < 16; ++r) {
        int e = ebase + r;
        float4 v = make_float4(0.f, 0.f, 0.f, 0.f);
        if (e < n_edges) {
            int src = edges[e * 3 + 0];
            int rel = edges[e * 3 + 1];
            float4 a = *(const float4*)(nodes + (size_t)src * DIM + lane * 4);
            float4 b = *(const float4*)(rels  + (size_t)rel * DIM + lane * 4);
            v.x = a.x + b.x; v.y = a.y + b.y; v.z = a.z + b.z; v.w = a.w + b.w;
        }
        *(float4*)(&xs[wave][r][lane * 4]) = v;
    }
    if (lane < 16) {
        int e = ebase + lane;
        dsts[wave][lane] = (e < n_edges) ? edges[e * 3 + 2] : -1;
    }
    // wave-private LDS slice: no workgroup barrier needed.

    // ---- GEMM: 16x128 (A) x 128x128 (W_agg^T) with f32 WMMA --------------
    // A frag (16x4 f32): lane holds A[l15][k0 + 2*half + j], j = 0..1
    // B frag (4x16 f32): lane holds B[k0 + 2*half + j][l15] = W_agg[n*16+l15][k]
    v8f acc[8] = {};
    const int koff = 2 * half;
    const float* wrow[8];
#pragma unroll
    for (int n = 0; n < 8; ++n) wrow[n] = W_agg + (size_t)(n * 16 + l15) * DIM;

    for (int k0 = 0; k0 < DIM; k0 += 4) {
        v2f a = *(const v2f*)(&xs[wave][l15][k0 + koff]);
#pragma unroll
        for (int n = 0; n < 8; ++n) {
            v2f b = *(const v2f*)(wrow[n] + k0 + koff);
            acc[n] = __builtin_amdgcn_wmma_f32_16x16x4_f32(
                false, a, false, b, (short)0, acc[n], false, false);
        }
    }

    // ---- scatter: C/D layout -> M = 8*half + i, N = n*16 + l15 -----------
#pragma unroll
    for (int i = 0; i < 8; ++i) {
        int row   = half * 8 + i;
        int dnode = dsts[wave][row];
        if (dnode >= 0) {
            float* outp = summed + (size_t)dnode * DIM + l15;
#pragma unroll
            for (int n = 0; n < 8; ++n)
                atomicAdd(outp + n * 16, acc[n][i]);
        }
    }
    if (lane < 16 && dsts[wave][lane] >= 0)
        atomicAdd(count + dsts[wave][lane], 1.0f);
}

// ---------------------------------------------------------------------------
// One wave handles 16 nodes: h = nodes @ W_self^T, then
// out = rrelu(h + summed/max(count,1)).
// ---------------------------------------------------------------------------
__global__ __launch_bounds__(128) void rgcn_node_kernel(
        const float* __restrict__ nodes,    // n_nodes x 128
        const float* __restrict__ W_self,   // 128 x 128
        const float* __restrict__ summed,   // n_nodes x 128
        const float* __restrict__ count,    // n_nodes
        float* __restrict__ out,            // n_nodes x 128
        int n_nodes) {
    __shared__ float xs[4][16][DIM];

    const int wave = threadIdx.x >> 5;
    const int lane = threadIdx.x & 31;
    const int l15  = lane & 15;
    const int half = lane >> 4;

    const int nbase = (blockIdx.x * 4 + wave) * 16;
    if (nbase >= n_nodes) return;

    for (int r = 0; r < 16; ++r) {
        float4 v = make_float4(0.f, 0.f, 0.f, 0.f);
        if (nbase + r < n_nodes)
            v = *(const float4*)(nodes + (size_t)(nbase + r) * DIM + lane * 4);
        *(float4*)(&xs[wave][r][lane * 4]) = v;
    }

    v8f acc[8] = {};
    const int koff = 2 * half;
    const float* wrow[8];
#pragma unroll
    for (int n = 0; n < 8; ++n) wrow[n] = W_self + (size_t)(n * 16 + l15) * DIM;

    for (int k0 = 0; k0 < DIM; k0 += 4) {
        v2f a = *(const v2f*)(&xs[wave][l15][k0 + koff]);
#pragma unroll
        for (int n = 0; n < 8; ++n) {
            v2f b = *(const v2f*)(wrow[n] + k0 + koff);
            acc[n] = __builtin_amdgcn_wmma_f32_16x16x4_f32(
                false, a, false, b, (short)0, acc[n], false, false);
        }
    }

#pragma unroll
    for (int i = 0; i < 8; ++i) {
        int node = nbase + half * 8 + i;
        if (node < n_nodes) {
            float c     = count[node];
            float scale = 1.0f / fmaxf(c, 1.0f);
            size_t base = (size_t)node * DIM + l15;
#pragma unroll
            for (int n = 0; n < 8; ++n) {
                float h = acc[n][i] + summed[base + n * 16] * scale;
                out[base + n * 16] = (h >= 0.0f) ? h : RRELU_SLOPE * h;
            }
        }
    }
}

// ---------------------------------------------------------------------------
extern "C" void kernel_launch(void* const* d_in, const int* in_sizes, int n_in,
                              void* d_out, int out_size, void* d_ws, size_t ws_size,
                              hipStream_t stream) {
    const float* nodes  = (const float*)d_in[0];   // nodes_embed (n_nodes x 128)
    const float* rels   = (const float*)d_in[1];   // edges_embed (n_rel x 128)
    const float* W_self = (const float*)d_in[2];   // 128 x 128
    const float* W_agg  = (const float*)d_in[3];   // 128 x 128
    const int*   edges  = (const int*)d_in[4];     // n_edges x 3

    const int n_nodes = in_sizes[0] / DIM;
    const int n_edges = in_sizes[4] / 3;

    float* out    = (float*)d_out;
    float* summed = (float*)d_ws;                        // n_nodes x 128
    float* count  = summed + (size_t)n_nodes * DIM;      // n_nodes

    // Re-zero accumulators every call (deterministic, graph-replay safe).
    long zn = (long)n_nodes * DIM + n_nodes;
    rgcn_zero_ws<<<2048, 256, 0, stream>>>(summed, zn);

    int etiles = (n_edges + 15) / 16;   // 16 edges per wave, 4 waves per block
    rgcn_edge_kernel<<<(etiles + 3) / 4, 128, 0, stream>>>(
        nodes, rels, W_agg, edges, summed, count, n_edges);

    int ntiles = (n_nodes + 15) / 16;   // 16 nodes per wave, 4 waves per block
    rgcn_node_kernel<<<(ntiles + 3) / 4, 128, 0, stream>>>(
        nodes, W_self, summed, count, out, n_nodes);
}